// FrameAttention_44418551776062
// MI455X (gfx1250) — compile-verified
//
#include <hip/hip_runtime.h>

// ---------------------------------------------------------------------------
// FrameAttention for MI455X (gfx1250, wave32, WMMA, async global->LDS).
// All GEMM work uses V_WMMA_F32_16X16X4_F32 (full fp32 precision).
// ---------------------------------------------------------------------------

typedef __attribute__((ext_vector_type(2))) float v2f;
typedef __attribute__((ext_vector_type(8))) float v8f;

static constexpr int B_  = 4;
static constexpr int C_  = 64;
static constexpr int T_  = 32;
static constexpr int SS_ = 512;
static constexpr int S_  = 32 * 32 * 32;   // H*W*T = 32768
static constexpr int OC_ = 3 * C_;         // 192 qkv channels
static constexpr int WP_ = 68;             // padded LDS row stride (bank-conflict-free)

__device__ __forceinline__ v8f wmma_f32_4(v2f a, v2f b, v8f c) {
  // D = A(16x4 f32) * B(4x16 f32) + C(16x16 f32)
  return __builtin_amdgcn_wmma_f32_16x16x4_f32(
      /*neg_a=*/false, a, /*neg_b=*/false, b,
      /*c_mod=*/(short)0, c, /*reuse_a=*/false, /*reuse_b=*/false);
}

// Async global->LDS copy (ASYNCcnt-tracked, no VGPR round trip).
__device__ __forceinline__ void lds_async_b128(unsigned lds_byte_off,
                                               const float* gaddr) {
  asm volatile("global_load_async_to_lds_b128 %0, %1, off"
               :: "v"(lds_byte_off), "v"(gaddr) : "memory");
}
__device__ __forceinline__ void wait_async0() {
  asm volatile("s_wait_asynccnt 0x0" ::: "memory");
}

// --------------------------- kernel 1: styles ------------------------------
__global__ void __launch_bounds__(256)
k_style(const float* __restrict__ style,
        const float* __restrict__ qkv_lw, const float* __restrict__ qkv_lb,
        const float* __restrict__ proj_lw, const float* __restrict__ proj_lb,
        float* __restrict__ s_qkv, float* __restrict__ s_proj) {
  int j = blockIdx.x * blockDim.x + threadIdx.x;       // 0..511
  if (j >= 2 * B_ * C_) return;
  bool is_proj = j >= B_ * C_;
  int jj = is_proj ? (j - B_ * C_) : j;
  int b = jj >> 6, c = jj & 63;
  const float* lw = (is_proj ? proj_lw : qkv_lw) + (size_t)c * SS_;
  const float* st = style + (size_t)b * SS_;
  float acc = 0.f;
#pragma unroll 8
  for (int k = 0; k < SS_; ++k) acc += st[k] * lw[k];
  acc += (is_proj ? proj_lb : qkv_lb)[c];
  (is_proj ? s_proj : s_qkv)[b * C_ + c] = acc;
}

// ------------------- kernel 2: modulate + demod-normalize ------------------
__global__ void __launch_bounds__(256)
k_mod(const float* __restrict__ qkv_w, const float* __restrict__ proj_w,
      const float* __restrict__ s_qkv, const float* __restrict__ s_proj,
      float* __restrict__ Wqkv, float* __restrict__ Wproj) {
  int j = blockIdx.x * blockDim.x + threadIdx.x;       // 0..1023
  const int nq = B_ * OC_;                             // 768 qkv rows
  if (j >= nq + B_ * C_) return;
  const float* wr; const float* sr; float* dst;
  if (j < nq) { int o = j % OC_, b = j / OC_;
    wr = qkv_w + (size_t)o * C_; sr = s_qkv + b * C_; dst = Wqkv + (size_t)j * C_; }
  else { int jj = j - nq; int o = jj % C_, b = jj / C_;
    wr = proj_w + (size_t)o * C_; sr = s_proj + b * C_; dst = Wproj + (size_t)jj * C_; }
  float acc = 1e-8f;
#pragma unroll 8
  for (int c = 0; c < C_; ++c) { float wm = wr[c] * sr[c]; acc += wm * wm; }
  float sig = rsqrtf(acc);
#pragma unroll 8
  for (int c = 0; c < C_; ++c) dst[c] = wr[c] * sr[c] * sig;
}

// ----------------------- kernel 3: qkv WMMA GEMM ---------------------------
// Each wave owns one 16-wide s-stripe and ALL 12 o-tiles (12 v8f accumulators).
// W[b] (192x64) staged once per block into LDS via async b128 copies; B
// fragment loaded once per K-step and reused by 12 WMMAs -> x read once total.
__global__ void __launch_bounds__(256)
k_qkv(const float* __restrict__ x, const float* __restrict__ Wqkv,
      const float* __restrict__ np, const float* __restrict__ bp,
      const float* __restrict__ noise, float* __restrict__ qkv) {
  __shared__ float smW[OC_ * WP_];                     // 192 rows, padded: ~51 KB
  int tid  = threadIdx.x;
  int lane = tid & 31;
  int wave = tid >> 5;
  int b    = blockIdx.x >> 8;                          // 1024 blocks
  int sgrp = blockIdx.x & 255;

  // stage W[b] (rows of 64 floats -> padded rows of 68) with async-to-LDS
  const float*  wsrc = Wqkv + (size_t)b * OC_ * C_;
  unsigned base = (unsigned)(size_t)&smW[0];
  for (int idx = tid; idx < OC_ * 16; idx += 256) {    // 3072 16B segments
    int row = idx >> 4, seg = idx & 15;
    lds_async_b128(base + (unsigned)((row * WP_ + seg * 4) * 4),
                   wsrc + row * C_ + seg * 4);
  }
  wait_async0();
  __syncthreads();

  int stile = sgrp * 8 + wave;                         // 0..2047
  int s0    = stile * 16;
  int mrow  = lane & 15;
  int khalf = (lane >> 4) * 2;                         // 0 | 2
  const float* xb = x + (size_t)b * C_ * S_ + s0 + mrow;

  v8f acc[12] = {};
#pragma unroll 4
  for (int k0 = 0; k0 < C_; k0 += 4) {
    const float* xp = xb + (size_t)(k0 + khalf) * S_;
    __builtin_prefetch(xp + (size_t)4 * S_, 0, 1);     // global_prefetch next slab
    v2f bb;
    bb.x = xp[0];
    bb.y = xp[S_];
#pragma unroll
    for (int ot = 0; ot < 12; ++ot) {
      const float* wp = &smW[(ot * 16 + mrow) * WP_ + k0 + khalf];
      v2f a; a.x = wp[0]; a.y = wp[1];                 // ds_load_b64, no conflicts
      acc[ot] = wmma_f32_4(a, bb, acc[ot]);
    }
  }

  int mofs = (lane >> 4) * 8;
  int s = s0 + mrow;
  float nz = noise[(size_t)b * S_ + s];
#pragma unroll
  for (int ot = 0; ot < 12; ++ot) {
#pragma unroll
    for (int r = 0; r < 8; ++r) {
      int o = ot * 16 + r + mofs;
      qkv[((size_t)(b * OC_ + o)) * S_ + s] = acc[ot][r] + np[o] * nz + bp[o];
    }
  }
}

// -------------- kernel 4: scores (Q.K^T, K=8192) + softmax -----------------
__global__ void __launch_bounds__(256)
k_scores(const float* __restrict__ qkv, float* __restrict__ attn) {
  __shared__ float sc[32 * 32];
  int lane = threadIdx.x & 31;
  int wave = threadIdx.x >> 5;
  int bn = blockIdx.x;                                 // 0..31
  int b = bn >> 3, n = bn & 7;
  for (int i = threadIdx.x; i < 1024; i += 256) sc[i] = 0.f;
  __syncthreads();

  int cch = n * 8 + wave;                              // channel handled by wave
  const float* qb = qkv + ((size_t)(b * OC_) + cch) * S_;
  const float* kb = qkv + ((size_t)(b * OC_) + C_ + cch) * S_;
  int khalf = (lane >> 4) * 2;
  int ml = lane & 15;
  v8f c00 = {}, c01 = {}, c10 = {}, c11 = {};
#pragma unroll 2
  for (int k0 = 0; k0 < 1024; k0 += 4) {               // hw contraction
    int ra = (k0 + khalf) * T_;
    v2f a0, a1, b0, b1;
    a0.x = qb[ra + ml];        a0.y = qb[ra + T_ + ml];
    a1.x = qb[ra + 16 + ml];   a1.y = qb[ra + T_ + 16 + ml];
    b0.x = kb[ra + ml];        b0.y = kb[ra + T_ + ml];
    b1.x = kb[ra + 16 + ml];   b1.y = kb[ra + T_ + 16 + ml];
    c00 = wmma_f32_4(a0, b0, c00);
    c01 = wmma_f32_4(a0, b1, c01);
    c10 = wmma_f32_4(a1, b0, c10);
    c11 = wmma_f32_4(a1, b1, c11);
  }
  int mofs = (lane >> 4) * 8;
#pragma unroll
  for (int r = 0; r < 8; ++r) {                        // ds_add_f32 reduce
    int i0 = r + mofs;
    atomicAdd(&sc[i0 * 32 + ml],             c00[r]);
    atomicAdd(&sc[i0 * 32 + 16 + ml],        c01[r]);
    atomicAdd(&sc[(i0 + 16) * 32 + ml],      c10[r]);
    atomicAdd(&sc[(i0 + 16) * 32 + 16 + ml], c11[r]);
  }
  __syncthreads();

  if (threadIdx.x < 32) {                              // softmax, one row / lane
    int i = threadIdx.x;
    float slope = exp2f(-(float)(n + 1));              // ALiBi slopes, NH=8
    const float scale = 0.011048543456039805f;         // 1/sqrt(8192)
    float row[32];
    float mx = -3.4e38f;
#pragma unroll
    for (int j = 0; j < 32; ++j) {
      float v = (j <= i) ? -1e18f                      // tril (incl diag) masked
                         : sc[i * 32 + j] * scale - slope * fabsf((float)(i - j));
      row[j] = v; mx = fmaxf(mx, v);
    }
    float sum = 0.f;
#pragma unroll
    for (int j = 0; j < 32; ++j) { float e = __expf(row[j] - mx); row[j] = e; sum += e; }
    float inv = 1.f / sum;
    float* ap = attn + (size_t)bn * 1024 + i * 32;
#pragma unroll
    for (int j = 0; j < 32; ++j) ap[j] = row[j] * inv;
  }
}

// ---------------------- kernel 5: out = attn @ V ---------------------------
__global__ void __launch_bounds__(256)
k_av(const float* __restrict__ qkv, const float* __restrict__ attn,
     float* __restrict__ aout) {
  int lane = threadIdx.x & 31;
  int wave = threadIdx.x >> 5;
  int gw = blockIdx.x * 8 + wave;                      // 16384 tile-waves
  int dt = gw & 511;
  int bn = gw >> 9;
  int b = bn >> 3, n = bn & 7;
  int d0  = dt * 16;
  int cp  = d0 >> 10;                                  // channel within head
  int hw0 = d0 & 1023;
  const float* vb = qkv + ((size_t)(b * OC_) + 2 * C_ + n * 8 + cp) * S_;
  const float* ap = attn + (size_t)bn * 1024;
  int khalf = (lane >> 4) * 2;
  int ml = lane & 15;
  v8f c0 = {}, c1 = {};
#pragma unroll
  for (int k0 = 0; k0 < 32; k0 += 4) {                 // K = T = 32
    v2f a0, a1, bb;
    a0.x = ap[ml * 32 + k0 + khalf];
    a0.y = ap[ml * 32 + k0 + khalf + 1];
    a1.x = ap[(16 + ml) * 32 + k0 + khalf];
    a1.y = ap[(16 + ml) * 32 + k0 + khalf + 1];
    bb.x = vb[(size_t)(hw0 + ml) * T_ + k0 + khalf];
    bb.y = vb[(size_t)(hw0 + ml) * T_ + k0 + khalf + 1];
    c0 = wmma_f32_4(a0, bb, c0);
    c1 = wmma_f32_4(a1, bb, c1);
  }
  int mofs = (lane >> 4) * 8;
  int cc = n * 8 + cp;
  float* ob = aout + ((size_t)(b * C_ + cc)) * S_;
#pragma unroll
  for (int r = 0; r < 8; ++r) {
    int i0 = r + mofs;
    ob[(size_t)(hw0 + ml) * T_ + i0]      = c0[r];     // tokens 0..15
    ob[(size_t)(hw0 + ml) * T_ + 16 + i0] = c1[r];     // tokens 16..31
  }
}

// ----------------------- kernel 6: proj WMMA GEMM --------------------------
// Same stripe scheme as k_qkv: 4 o-tiles per wave, W staged in LDS (async).
__global__ void __launch_bounds__(256)
k_proj(const float* __restrict__ ain, const float* __restrict__ Wp,
       const float* __restrict__ np, const float* __restrict__ bp,
       const float* __restrict__ noise, float* __restrict__ out) {
  __shared__ float smW[C_ * WP_];                      // 64 padded rows: ~17 KB
  int tid  = threadIdx.x;
  int lane = tid & 31;
  int wave = tid >> 5;
  int b    = blockIdx.x >> 8;                          // 1024 blocks
  int sgrp = blockIdx.x & 255;

  const float* wsrc = Wp + (size_t)b * C_ * C_;
  unsigned base = (unsigned)(size_t)&smW[0];
  for (int idx = tid; idx < C_ * 16; idx += 256) {     // 1024 16B segments
    int row = idx >> 4, seg = idx & 15;
    lds_async_b128(base + (unsigned)((row * WP_ + seg * 4) * 4),
                   wsrc + row * C_ + seg * 4);
  }
  wait_async0();
  __syncthreads();

  int stile = sgrp * 8 + wave;
  int s0    = stile * 16;
  int mrow  = lane & 15;
  int khalf = (lane >> 4) * 2;
  const float* xb = ain + (size_t)b * C_ * S_ + s0 + mrow;

  v8f acc[4] = {};
#pragma unroll 4
  for (int k0 = 0; k0 < C_; k0 += 4) {
    const float* xp = xb + (size_t)(k0 + khalf) * S_;
    v2f bb;
    bb.x = xp[0];
    bb.y = xp[S_];
#pragma unroll
    for (int ot = 0; ot < 4; ++ot) {
      const float* wp = &smW[(ot * 16 + mrow) * WP_ + k0 + khalf];
      v2f a; a.x = wp[0]; a.y = wp[1];
      acc[ot] = wmma_f32_4(a, bb, acc[ot]);
    }
  }

  int mofs = (lane >> 4) * 8;
  int s = s0 + mrow;
  float nz = noise[(size_t)b * S_ + s];
#pragma unroll
  for (int ot = 0; ot < 4; ++ot) {
#pragma unroll
    for (int r = 0; r < 8; ++r) {
      int o = ot * 16 + r + mofs;
      out[((size_t)(b * C_ + o)) * S_ + s] = acc[ot][r] + np[o] * nz + bp[o];
    }
  }
}

// ---------------------------------------------------------------------------
extern "C" void kernel_launch(void* const* d_in, const int* in_sizes, int n_in,
                              void* d_out, int out_size, void* d_ws, size_t ws_size,
                              hipStream_t stream) {
  const float* x          = (const float*)d_in[0];
  const float* style      = (const float*)d_in[1];
  const float* qkv_w      = (const float*)d_in[2];
  const float* qkv_lw     = (const float*)d_in[3];
  const float* qkv_lb     = (const float*)d_in[4];
  const float* qkv_np     = (const float*)d_in[5];
  const float* qkv_bp     = (const float*)d_in[6];
  const float* qkv_noise  = (const float*)d_in[7];
  const float* proj_w     = (const float*)d_in[8];
  const float* proj_lw    = (const float*)d_in[9];
  const float* proj_lb    = (const float*)d_in[10];
  const float* proj_np    = (const float*)d_in[11];
  const float* proj_bp    = (const float*)d_in[12];
  const float* proj_noise = (const float*)d_in[13];
  float* out = (float*)d_out;

  // Workspace layout (floats). Total ~128.4 MiB.
  float* ws      = (float*)d_ws;
  float* s_qkv   = ws;                                  // 256
  float* s_proj  = ws + 256;                            // 256
  float* Wqkv    = ws + 512;                            // 49152
  float* Wproj   = ws + 49664;                          // 16384
  float* qkv     = ws + 66048;                          // 25165824 (96 MiB)
  float* attn    = ws + 25231872;                       // 32768
  float* aout    = ws + 25264640;                       // 8388608 (32 MiB)

  k_style<<<2, 256, 0, stream>>>(style, qkv_lw, qkv_lb, proj_lw, proj_lb,
                                 s_qkv, s_proj);
  k_mod<<<4, 256, 0, stream>>>(qkv_w, proj_w, s_qkv, s_proj, Wqkv, Wproj);
  k_qkv<<<1024, 256, 0, stream>>>(x, Wqkv, qkv_np, qkv_bp, qkv_noise, qkv);
  k_scores<<<32, 256, 0, stream>>>(qkv, attn);
  k_av<<<2048, 256, 0, stream>>>(qkv, attn, aout);
  k_proj<<<1024, 256, 0, stream>>>(aout, Wproj, proj_np, proj_bp, proj_noise, out);
}